// FasterRCNNLoss_24979529793814
// MI455X (gfx1250) — compile-verified
//
#include <hip/hip_runtime.h>
#include <hip/hip_bf16.h>
#include <math.h>

typedef unsigned long long ull;
typedef __attribute__((ext_vector_type(2))) float v2f;
typedef __attribute__((ext_vector_type(8))) float v8f;

#define G 32
#define RPN_BATCH 256
#define RPN_HALF 128
#define ROI_BATCH 128
#define ROI_POS 32

// ---------------- math helpers (match reference exactly) ----------------

__device__ __forceinline__ float box_iou(float a0, float a1, float a2, float a3,
                                         float b0, float b1, float b2, float b3) {
  float lx = fmaxf(a0, b0), ly = fmaxf(a1, b1);
  float rx = fminf(a2, b2), ry = fminf(a3, b3);
  float w = fmaxf(rx - lx, 0.f), h = fmaxf(ry - ly, 0.f);
  float inter = w * h;
  float aa = (a2 - a0) * (a3 - a1);
  float ab = (b2 - b0) * (b3 - b1);
  return inter / (aa + ab - inter + 1e-6f);
}

__device__ __forceinline__ float smooth_l1(float x) {
  float ax = fabsf(x);
  return ax < 1.f ? 0.5f * x * x : ax - 0.5f;
}

__device__ __forceinline__ void encode4(const float* a, const float* t, float* o) {
  float wa = a[2] - a[0], ha = a[3] - a[1];
  float xa = a[0] + 0.5f * wa, ya = a[1] + 0.5f * ha;
  float wt = t[2] - t[0], ht = t[3] - t[1];
  float xt = t[0] + 0.5f * wt, yt = t[1] + 0.5f * ht;
  o[0] = (xt - xa) / wa;
  o[1] = (yt - ya) / ha;
  o[2] = logf(wt / wa + 1e-6f);
  o[3] = logf(ht / ha + 1e-6f);
}

// ---------------- init: zero argmax keys + zero (padded) partial-sum buffers ----------------

__global__ void init_kernel(ull* gtbest, int ng, float* partials, int nf) {
  int i = blockIdx.x * blockDim.x + threadIdx.x;
  if (i < ng) gtbest[i] = 0ull;
  if (i < nf) partials[i] = 0.f;
}

// ---------------- RPN pass 1: per-anchor max IoU / argmax + per-GT column argmax ----------------

__global__ void rpn_iou_kernel(const float* __restrict__ anchors,
                               const float* __restrict__ gt,   // B*G*4
                               float* __restrict__ maxiou,     // B*N
                               int* __restrict__ amax,         // B*N
                               ull* __restrict__ gtbest,       // B*G, packed keys
                               int N) {
  __shared__ float sgt[G * 4];
  __shared__ ull scol[G];
  int t = threadIdx.x;
  int b = blockIdx.y;
  if (t < G * 4) sgt[t] = gt[b * G * 4 + t];
  if (t < G) scol[t] = 0ull;
  __syncthreads();

  int i = blockIdx.x * blockDim.x + t;
  bool act = i < N;
  float a0 = 0.f, a1 = 0.f, a2 = 1.f, a3 = 1.f;
  if (act) {
    a0 = anchors[(size_t)i * 4 + 0];
    a1 = anchors[(size_t)i * 4 + 1];
    a2 = anchors[(size_t)i * 4 + 2];
    a3 = anchors[(size_t)i * 4 + 3];
  }
  float best = -1.f;
  int bj = 0;
  int lane = t & 31;
  for (int j = 0; j < G; ++j) {
    float iou = box_iou(a0, a1, a2, a3,
                        sgt[j * 4 + 0], sgt[j * 4 + 1], sgt[j * 4 + 2], sgt[j * 4 + 3]);
    if (iou > best) { best = iou; bj = j; }
    // packed key: max iou wins; ties -> smallest anchor index (matches jnp.argmax axis=0)
    ull key = act ? ((((ull)__float_as_uint(iou)) << 32) |
                    (ull)(0x7FFFFFFFu - (unsigned)i))
                  : 0ull;
    for (int off = 16; off > 0; off >>= 1) {
      ull o = __shfl_xor(key, off, 32);
      if (o > key) key = o;
    }
    if (lane == 0) atomicMax(&scol[j], key);
  }
  if (act) {
    maxiou[(size_t)b * N + i] = best;
    amax[(size_t)b * N + i] = bj;
  }
  __syncthreads();
  if (t < G) atomicMax(&gtbest[b * G + t], scol[t]);  // u64 max: order-independent -> deterministic
}

// ---------------- RPN pass 2: per-block pos/neg counts ----------------

__global__ void rpn_count_kernel(const float* __restrict__ maxiou,
                                 const ull* __restrict__ gtbest,
                                 int* __restrict__ blkpos, int* __restrict__ blkneg,
                                 int N, int NB) {
  __shared__ int sidx[G];
  __shared__ int wp[8], wn[8];
  int t = threadIdx.x, b = blockIdx.y, blk = blockIdx.x;
  if (t < G)
    sidx[t] = (int)(0x7FFFFFFFu - (unsigned)(gtbest[b * G + t] & 0xFFFFFFFFull));
  __syncthreads();
  int i = blk * blockDim.x + t;
  bool act = i < N;
  bool member = false;
  if (act) {
#pragma unroll
    for (int j = 0; j < G; ++j) member |= (i == sidx[j]);
  }
  float mi = act ? maxiou[(size_t)b * N + i] : 0.5f;
  bool pos = act && (mi > 0.7f || member);
  bool neg = act && (mi < 0.3f) && !member;
  ull bp = __ballot(pos ? 1 : 0), bn = __ballot(neg ? 1 : 0);
  int lane = t & 31, wid = t >> 5;
  if (lane == 0) { wp[wid] = __popcll(bp); wn[wid] = __popcll(bn); }
  __syncthreads();
  if (t == 0) {
    int sp = 0, sn = 0;
    for (int w = 0; w < 8; ++w) { sp += wp[w]; sn += wn[w]; }
    blkpos[b * NB + blk] = sp;
    blkneg[b * NB + blk] = sn;
  }
}

// ---------------- RPN pass 3: scan block counts (NB <= 1024) + totals/meta ----------------

__global__ void rpn_scan_kernel(const int* __restrict__ blkpos, const int* __restrict__ blkneg,
                                int* __restrict__ offpos, int* __restrict__ offneg,
                                int* __restrict__ meta, int NB) {
  __shared__ int sp[1024], sn[1024];
  int t = threadIdx.x, b = blockIdx.x;
  int cp = (t < NB) ? blkpos[b * NB + t] : 0;
  int cn = (t < NB) ? blkneg[b * NB + t] : 0;
  sp[t] = cp; sn[t] = cn;
  __syncthreads();
  for (int off = 1; off < 1024; off <<= 1) {
    int ap = 0, an = 0;
    if (t >= off) { ap = sp[t - off]; an = sn[t - off]; }
    __syncthreads();
    sp[t] += ap; sn[t] += an;
    __syncthreads();
  }
  if (t < NB) { offpos[b * NB + t] = sp[t] - cp; offneg[b * NB + t] = sn[t] - cn; }
  if (t == 0) {
    int pt = sp[NB - 1], nt = sn[NB - 1];
    int np = pt < RPN_HALF ? pt : RPN_HALF;       // n_pos after pos truncation
    int quota = RPN_BATCH - np;                    // negative quota
    int nn = nt < quota ? nt : quota;
    int* m = meta + b * 8;
    m[0] = pt; m[1] = nt; m[2] = np; m[3] = quota; m[4] = np + nn;  // mask count
  }
}

// ---------------- RPN pass 4: ranks -> labels -> BCE + smooth-L1 partials ----------------

__global__ void rpn_loss_kernel(const float* __restrict__ anchors, const float* __restrict__ gt,
                                const float* __restrict__ scores, const float* __restrict__ deltas,
                                const float* __restrict__ maxiou, const int* __restrict__ amax,
                                const ull* __restrict__ gtbest,
                                const int* __restrict__ offpos, const int* __restrict__ offneg,
                                const int* __restrict__ meta,
                                float* __restrict__ blkcls, float* __restrict__ blkreg,
                                int N, int NB, int NBS) {
  __shared__ float sgt[G * 4];
  __shared__ int sidx[G];
  __shared__ int wp[8], wn[8], wpx[8], wnx[8];
  __shared__ float rcls[256], rreg[256];
  int t = threadIdx.x, b = blockIdx.y, blk = blockIdx.x;
  if (t < G * 4) sgt[t] = gt[b * G * 4 + t];
  if (t < G)
    sidx[t] = (int)(0x7FFFFFFFu - (unsigned)(gtbest[b * G + t] & 0xFFFFFFFFull));
  __syncthreads();

  int i = blk * 256 + t;
  bool act = i < N;
  bool member = false;
  if (act) {
#pragma unroll
    for (int j = 0; j < G; ++j) member |= (i == sidx[j]);
  }
  float mi = act ? maxiou[(size_t)b * N + i] : 0.5f;
  bool pos = act && (mi > 0.7f || member);
  bool neg = act && (mi < 0.3f) && !member;
  ull bp = __ballot(pos ? 1 : 0), bn = __ballot(neg ? 1 : 0);
  int lane = t & 31, wid = t >> 5;
  if (lane == 0) { wp[wid] = __popcll(bp); wn[wid] = __popcll(bn); }
  __syncthreads();
  if (t == 0) {
    int s = 0;
    for (int w = 0; w < 8; ++w) { wpx[w] = s; s += wp[w]; }
    s = 0;
    for (int w = 0; w < 8; ++w) { wnx[w] = s; s += wn[w]; }
  }
  __syncthreads();

  int quota = meta[b * 8 + 3];
  int label = -1;
  ull lmask = (1ull << lane) - 1ull;
  if (pos) {
    int rank = offpos[b * NB + blk] + wpx[wid] + __popcll(bp & lmask) + 1;
    if (rank <= RPN_HALF) label = 1;
  } else if (neg) {
    int rank = offneg[b * NB + blk] + wnx[wid] + __popcll(bn & lmask) + 1;
    if (rank <= quota) label = 0;
  }

  float clsv = 0.f, regv = 0.f;
  if (label >= 0) {
    float l = scores[(size_t)b * N + i];
    float tt = (float)label;
    clsv = fmaxf(l, 0.f) - l * tt + log1pf(expf(-fabsf(l)));
  }
  if (label == 1) {
    float a[4] = {anchors[(size_t)i * 4 + 0], anchors[(size_t)i * 4 + 1],
                  anchors[(size_t)i * 4 + 2], anchors[(size_t)i * 4 + 3]};
    int j = amax[(size_t)b * N + i];
    float tg[4];
    encode4(a, &sgt[j * 4], tg);
    const float* d = deltas + ((size_t)b * N + i) * 4;
    regv = smooth_l1(d[0] - tg[0]) + smooth_l1(d[1] - tg[1]) +
           smooth_l1(d[2] - tg[2]) + smooth_l1(d[3] - tg[3]);
  }
  rcls[t] = clsv; rreg[t] = regv;
  __syncthreads();
  for (int off = 128; off > 0; off >>= 1) {
    if (t < off) { rcls[t] += rcls[t + off]; rreg[t] += rreg[t + off]; }
    __syncthreads();
  }
  if (t == 0) { blkcls[b * NBS + blk] = rcls[0]; blkreg[b * NBS + blk] = rreg[0]; }
}

// ---------------- Detection head loss: one 1024-thread workgroup per batch ----------------

__global__ void __launch_bounds__(1024)
det_loss_kernel(const float* __restrict__ cls, const float* __restrict__ bdel,
                const float* __restrict__ proposals, const float* __restrict__ gt,
                const int* __restrict__ gtl, float* __restrict__ det, int P, int C) {
  __shared__ float sgt[G * 4];
  __shared__ int sgl[G];
  __shared__ int sassign[1024];
  __shared__ int sargj[1024];
  __shared__ int sp[1024], sn[1024];
  __shared__ int samp[ROI_BATCH];
  __shared__ float rc[ROI_BATCH], rr[ROI_BATCH];
  __shared__ int rn[ROI_BATCH];
  int t = threadIdx.x, b = blockIdx.x;
  if (t < G * 4) sgt[t] = gt[b * G * 4 + t];
  if (t < G) sgl[t] = gtl[b * G + t];
  __syncthreads();

  int pos = 0, neg = 0;
  if (t < P) {
    const float* pb = proposals + ((size_t)b * P + t) * 4;
    float p0 = pb[0], p1 = pb[1], p2 = pb[2], p3 = pb[3];
    float best = -1.f;
    int bj = 0;
    for (int j = 0; j < G; ++j) {
      float iou = box_iou(p0, p1, p2, p3,
                          sgt[j * 4 + 0], sgt[j * 4 + 1], sgt[j * 4 + 2], sgt[j * 4 + 3]);
      if (iou > best) { best = iou; bj = j; }
    }
    int asg = (best >= 0.5f) ? sgl[bj] : 0;
    sassign[t] = asg; sargj[t] = bj;
    pos = asg > 0 ? 1 : 0;
    neg = 1 - pos;
  } else {
    sassign[t] = 0; sargj[t] = 0;
  }
  sp[t] = pos; sn[t] = neg;
  __syncthreads();
  for (int off = 1; off < 1024; off <<= 1) {
    int ap = 0, an = 0;
    if (t >= off) { ap = sp[t - off]; an = sn[t - off]; }
    __syncthreads();
    sp[t] += ap; sn[t] += an;
    __syncthreads();
  }
  int pos_tot = sp[P - 1], neg_tot = sn[P - 1];
  int smin = pos_tot < ROI_POS ? pos_tot : ROI_POS;
  // argsort(key)[:128] analytically: [sel pos by rank] ++ [neg by rank] ++ [overflow pos by idx]
  if (t < P) {
    if (pos) {
      int r = sp[t] - 1;
      int slot = (r < ROI_POS) ? r : (smin + neg_tot + (r - ROI_POS));
      if (slot < ROI_BATCH) samp[slot] = t;
    } else {
      int slot = smin + (sn[t] - 1);
      if (slot < ROI_BATCH) samp[slot] = t;
    }
  }
  __syncthreads();

  if (t < ROI_BATCH) {
    int i = samp[t];
    int tl = sassign[i];
    const float* row = cls + ((size_t)b * P + i) * C;
    float mx = row[0];
    for (int c = 1; c < C; ++c) mx = fmaxf(mx, row[c]);
    float se = 0.f;
    for (int c = 0; c < C; ++c) se += expf(row[c] - mx);
    float lse = mx + logf(se);
    rc[t] = -(row[tl] - lse);
    float regm = 0.f;
    int pm = tl > 0 ? 1 : 0;
    if (pm) {
      const float* pb = proposals + ((size_t)b * P + i) * 4;
      float pbox[4] = {pb[0], pb[1], pb[2], pb[3]};
      float tg[4];
      encode4(pbox, &sgt[sargj[i] * 4], tg);
      const float* pd = bdel + ((size_t)b * P + i) * (size_t)(4 * C) + tl * 4;
      regm = smooth_l1(pd[0] - tg[0]) + smooth_l1(pd[1] - tg[1]) +
             smooth_l1(pd[2] - tg[2]) + smooth_l1(pd[3] - tg[3]);
    }
    rr[t] = regm; rn[t] = pm;
  }
  __syncthreads();
  for (int off = ROI_BATCH / 2; off > 0; off >>= 1) {
    if (t < off) { rc[t] += rc[t + off]; rr[t] += rr[t + off]; rn[t] += rn[t + off]; }
    __syncthreads();
  }
  if (t == 0) {
    float clsL = rc[0] / (float)ROI_BATCH;
    int np = rn[0];
    float regL = np > 0 ? rr[0] / fmaxf((float)(np * 4), 1.f) : 0.f;
    det[b] = clsL + regL;
  }
}

// ---------------- final reduce: WMMA ones-vector sum of zero-padded block partials ----------------

#if defined(__has_builtin)
#if __has_builtin(__builtin_amdgcn_wmma_f32_16x16x4_f32)
#define HAVE_WMMA_F32X4 1
#endif
#endif

// p points at an n64*64-element, zero-padded, 8-byte-aligned array.
__device__ __forceinline__ float wave_wmma_sum(const float* p, int n64, int lane) {
#ifdef HAVE_WMMA_F32X4
  // D = A(16x4) * ones(4x16) + C accumulates f32 row-sums of 64-element chunks.
  v2f ones; ones.x = 1.f; ones.y = 1.f;
  v8f acc = {};
  const v2f* pv = (const v2f*)p;  // unconditional global_load_b64 per lane per chunk
  for (int c = 0; c < n64; ++c) {
    v2f a = pv[c * 32 + lane];
    acc = __builtin_amdgcn_wmma_f32_16x16x4_f32(false, a, false, ones,
                                                (short)0, acc, false, false);
  }
  // lanes 0..15 hold C[0..7][lane], lanes 16..31 hold C[8..15][lane-16]; cols identical.
  float s = acc[0] + acc[1] + acc[2] + acc[3] + acc[4] + acc[5] + acc[6] + acc[7];
  s += __shfl_xor(s, 16, 32);
  return s;
#else
  int n = n64 * 64;
  float s = 0.f;
  for (int i = lane; i < n; i += 32) s += p[i];
  for (int off = 16; off > 0; off >>= 1) s += __shfl_xor(s, off, 32);
  return s;
#endif
}

__global__ void __launch_bounds__(32)
final_kernel(const float* __restrict__ blkcls, const float* __restrict__ blkreg,
             const int* __restrict__ meta, const float* __restrict__ det,
             float* __restrict__ out, int NBS, int B) {
  int lane = threadIdx.x;  // single full wave (32 threads): EXEC all-ones for WMMA
  int n64 = NBS >> 6;
  float total = 0.f;
  for (int b = 0; b < B; ++b) {
    float cs = wave_wmma_sum(blkcls + (size_t)b * NBS, n64, lane);
    float rs = wave_wmma_sum(blkreg + (size_t)b * NBS, n64, lane);
    int mc = meta[b * 8 + 4];
    int np = meta[b * 8 + 2];
    float clsL = cs / fmaxf((float)mc, 1.f);
    float regL = (np > 0) ? rs / fmaxf((float)(np * 4), 1.f) : 0.f;
    total += clsL + regL + det[b];
  }
  if (lane == 0) out[0] = total;
}

// ---------------- launch ----------------

static inline size_t align16(size_t x) { return (x + 15) & ~(size_t)15; }

extern "C" void kernel_launch(void* const* d_in, const int* in_sizes, int n_in,
                              void* d_out, int out_size, void* d_ws, size_t ws_size,
                              hipStream_t stream) {
  const float* rpn_scores  = (const float*)d_in[0];
  const float* rpn_deltas  = (const float*)d_in[1];
  const float* cls_scores  = (const float*)d_in[2];
  const float* bbox_deltas = (const float*)d_in[3];
  const float* anchors     = (const float*)d_in[4];
  const float* proposals   = (const float*)d_in[5];
  const float* gt_boxes    = (const float*)d_in[6];
  const int*   gt_labels   = (const int*)d_in[7];

  const int B   = in_sizes[6] / (G * 4);   // 4
  const int N   = in_sizes[4] / 4;         // 250000
  const int P   = in_sizes[5] / (B * 4);   // 1000
  const int C   = in_sizes[2] / (B * P);   // 81
  const int NB  = (N + 255) / 256;         // 977 (must be <= 1024 for the scan kernel)
  const int NBS = (NB + 63) & ~63;         // 1024: padded stride for WMMA reduce

  char* w = (char*)d_ws;
  float* ws_maxiou = (float*)w;  w += align16((size_t)B * N * sizeof(float));
  int*   ws_amax   = (int*)w;    w += align16((size_t)B * N * sizeof(int));
  ull*   ws_gtbest = (ull*)w;    w += align16((size_t)B * G * sizeof(ull));
  int*   ws_blkpos = (int*)w;    w += align16((size_t)B * NB * sizeof(int));
  int*   ws_blkneg = (int*)w;    w += align16((size_t)B * NB * sizeof(int));
  int*   ws_offpos = (int*)w;    w += align16((size_t)B * NB * sizeof(int));
  int*   ws_offneg = (int*)w;    w += align16((size_t)B * NB * sizeof(int));
  int*   ws_meta   = (int*)w;    w += align16((size_t)B * 8 * sizeof(int));
  float* ws_blkcls = (float*)w;  w += (size_t)B * NBS * sizeof(float);  // contiguous with blkreg
  float* ws_blkreg = (float*)w;  w += (size_t)B * NBS * sizeof(float);
  float* ws_det    = (float*)w;  w += align16((size_t)B * sizeof(float));

  const int nzero = 2 * B * NBS;  // zero both padded partial buffers each call
  init_kernel<<<(nzero + 255) / 256, 256, 0, stream>>>(ws_gtbest, B * G, ws_blkcls, nzero);
  rpn_iou_kernel<<<dim3(NB, B), 256, 0, stream>>>(anchors, gt_boxes, ws_maxiou, ws_amax,
                                                  ws_gtbest, N);
  rpn_count_kernel<<<dim3(NB, B), 256, 0, stream>>>(ws_maxiou, ws_gtbest,
                                                    ws_blkpos, ws_blkneg, N, NB);
  rpn_scan_kernel<<<B, 1024, 0, stream>>>(ws_blkpos, ws_blkneg, ws_offpos, ws_offneg,
                                          ws_meta, NB);
  rpn_loss_kernel<<<dim3(NB, B), 256, 0, stream>>>(anchors, gt_boxes, rpn_scores, rpn_deltas,
                                                   ws_maxiou, ws_amax, ws_gtbest,
                                                   ws_offpos, ws_offneg, ws_meta,
                                                   ws_blkcls, ws_blkreg, N, NB, NBS);
  det_loss_kernel<<<B, 1024, 0, stream>>>(cls_scores, bbox_deltas, proposals, gt_boxes,
                                          gt_labels, ws_det, P, C);
  final_kernel<<<1, 32, 0, stream>>>(ws_blkcls, ws_blkreg, ws_meta, ws_det,
                                     (float*)d_out, NBS, B);
}